// GeometryOptimalTransportSparseTopK_30958124270327
// MI455X (gfx1250) — compile-verified
//
#include <hip/hip_runtime.h>
#include <hip/hip_bf16.h>
#include <math.h>

// ---------------- problem constants (reference shapes) ----------------
constexpr int B_  = 4;
constexpr int N_  = 4096;   // sources
constexpr int M_  = 4096;   // targets
constexpr int D_  = 128;    // feature dim
constexpr int K_  = 64;     // topk support
constexpr int IT_ = 8;      // sinkhorn iters
constexpr float THRESH2 = 0.2f * 0.2f;            // 0.04
constexpr float INV_EPS = 1.0f / (0.01f + 1e-8f); // ~100
constexpr int BINS = 256;
constexpr float BIN_SCALE = (float)BINS / THRESH2; // 6400

typedef float v2f __attribute__((ext_vector_type(2)));
typedef float v8f __attribute__((ext_vector_type(8)));

// ===================================================================
// Kernel 1: per-16-target-row tile: WMMA f32 16x16x4 distance tiles,
// histogram-select top-64, emit (idx, log_k) sparse rows.
// dist^2 = [-2tx,-2ty,|t|^2,1] . [sx,sy,1,|s|^2]   (single K=4 WMMA)
// ===================================================================
__global__ __launch_bounds__(128) void k_topk(
    const float* __restrict__ src_locs,      // [B,N,2]
    const float* __restrict__ tgt_locs,      // [B,M,2]
    const unsigned char* __restrict__ smask, // [B,N]
    const unsigned char* __restrict__ tmask, // [B,M]
    int*   __restrict__ out_idx,             // [B*M,64]
    float* __restrict__ out_logk,            // [B*M,64]
    int*   __restrict__ out_cnt)             // [B*M]
{
    __shared__ unsigned int hist[16][BINS];      // 16 KB
    __shared__ float trow[16][3];                // tx, ty, |t|^2
    __shared__ unsigned char tval[16];
    __shared__ int   Tbin[16], TakeT[16];
    __shared__ unsigned int emit_cnt[16], tie_cnt[16];

    const int wg   = blockIdx.x;
    const int b    = wg / (M_ / 16);
    const int m0   = (wg % (M_ / 16)) * 16;
    const int tid  = threadIdx.x;
    const int wave = tid >> 5;
    const int lane = tid & 31;
    const int nlane = lane & 15;    // column within 16-wide chunk (D layout)

    for (int i = tid; i < 16 * BINS; i += 128) ((unsigned int*)hist)[i] = 0u;
    if (tid < 16) {
        float tx = tgt_locs[(b * M_ + m0 + tid) * 2 + 0];
        float ty = tgt_locs[(b * M_ + m0 + tid) * 2 + 1];
        trow[tid][0] = tx; trow[tid][1] = ty; trow[tid][2] = tx * tx + ty * ty;
        tval[tid] = tmask[b * M_ + m0 + tid];
    }
    __syncthreads();

    // Build A (16x4 f32): VGPR0 = K0|K2, VGPR1 = K1|K3 (ISA layout)
    v2f Amat;
    if (lane < 16) { Amat.x = -2.0f * trow[nlane][0]; Amat.y = -2.0f * trow[nlane][1]; }
    else           { Amat.x = trow[nlane][2];          Amat.y = 1.0f; }
    const int rbase = (lane >> 4) * 8;   // rows held by this lane in D (0..7 or 8..15)

    // Per-lane register mask of target-row validity (kills per-element LDS reads)
    unsigned int tvm = 0u;
#pragma unroll
    for (int g = 0; g < 8; ++g) tvm |= (tval[rbase + g] ? 1u : 0u) << g;

    const int steps = N_ / 64;           // 4 waves x 16 cols per step

    // ---------------- pass 1: histogram ----------------
    for (int step = 0; step < steps; ++step) {
        const int n = step * 64 + wave * 16 + nlane;
        float sx = src_locs[(b * N_ + n) * 2 + 0];
        float sy = src_locs[(b * N_ + n) * 2 + 1];
        v2f Bmat;
        if (lane < 16) { Bmat.x = sx; Bmat.y = sy; }
        else           { Bmat.x = 1.0f; Bmat.y = sx * sx + sy * sy; }
        v8f Cz = {};
        v8f Dv = __builtin_amdgcn_wmma_f32_16x16x4_f32(
            false, Amat, false, Bmat, (short)0, Cz, false, false);
        const bool sm = (smask[b * N_ + n] != 0);
#pragma unroll
        for (int g = 0; g < 8; ++g) {
            float d = Dv[g];
            if (sm && ((tvm >> g) & 1u) && d < THRESH2) {
                int bin = (int)(d * BIN_SCALE);
                bin = bin < 0 ? 0 : (bin > BINS - 1 ? BINS - 1 : bin);
                atomicAdd(&hist[rbase + g][bin], 1u);
            }
        }
    }
    __syncthreads();

    // ---------------- threshold per row ----------------
    if (tid < 16) {
        int cum = 0, t = BINS, take = 0;
        for (int bb = 0; bb < BINS; ++bb) {
            int c = (int)hist[tid][bb];
            if (cum + c > K_) { t = bb; take = K_ - cum; break; }
            cum += c;
        }
        Tbin[tid] = t; TakeT[tid] = take;
        emit_cnt[tid] = 0u; tie_cnt[tid] = 0u;
        out_cnt[b * M_ + m0 + tid] = (t == BINS) ? cum : K_;
    }
    __syncthreads();

    // Register-cache per-row threshold state for this lane's 8 rows
    int t_r[8], take_r[8];
#pragma unroll
    for (int g = 0; g < 8; ++g) { t_r[g] = Tbin[rbase + g]; take_r[g] = TakeT[rbase + g]; }

    // ---------------- pass 2: re-scan + emit ----------------
    for (int step = 0; step < steps; ++step) {
        const int n = step * 64 + wave * 16 + nlane;
        float sx = src_locs[(b * N_ + n) * 2 + 0];
        float sy = src_locs[(b * N_ + n) * 2 + 1];
        v2f Bmat;
        if (lane < 16) { Bmat.x = sx; Bmat.y = sy; }
        else           { Bmat.x = 1.0f; Bmat.y = sx * sx + sy * sy; }
        v8f Cz = {};
        v8f Dv = __builtin_amdgcn_wmma_f32_16x16x4_f32(
            false, Amat, false, Bmat, (short)0, Cz, false, false);
        const bool sm = (smask[b * N_ + n] != 0);
#pragma unroll
        for (int g = 0; g < 8; ++g) {
            float d = Dv[g];
            if (sm && ((tvm >> g) & 1u) && d < THRESH2) {
                int bin = (int)(d * BIN_SCALE);
                bin = bin < 0 ? 0 : (bin > BINS - 1 ? BINS - 1 : bin);
                const int r = rbase + g;
                int pos = -1;
                if (bin < t_r[g]) {
                    pos = (int)atomicAdd(&emit_cnt[r], 1u);
                } else if (bin == t_r[g]) {
                    unsigned int tt = atomicAdd(&tie_cnt[r], 1u);
                    if ((int)tt < take_r[g]) pos = (int)atomicAdd(&emit_cnt[r], 1u);
                }
                if (pos >= 0 && pos < K_) {
                    int base = (b * M_ + m0 + r) * K_;
                    out_idx[base + pos]  = n;
                    out_logk[base + pos] = -d * INV_EPS;
                }
            }
        }
    }
    __syncthreads();

    // pad unused slots: idx=0, logk=-1e30 (exp -> 0, branch-free downstream)
    if (tid < 16) {
        int base = (b * M_ + m0 + tid) * K_;
        for (int p = (int)emit_cnt[tid]; p < K_; ++p) {
            out_idx[base + p]  = 0;
            out_logk[base + p] = -1e30f;
        }
    }
}

// ---------------- init: zero v and colsum ----------------
__global__ __launch_bounds__(256) void k_init(float* __restrict__ v,
                                              float* __restrict__ col) {
    int i = blockIdx.x * 256 + threadIdx.x;
    if (i < B_ * N_) { v[i] = 0.0f; col[i] = 0.0f; }
}

// ---------------- Sinkhorn row update: u = -log sum_k exp(logk + v[idx]) ----
__global__ __launch_bounds__(256) void k_row(
    const int* __restrict__ idx, const float* __restrict__ logk,
    const float* __restrict__ v, float* __restrict__ u)
{
    int row  = blockIdx.x * 8 + (threadIdx.x >> 5);   // wave per row
    int lane = threadIdx.x & 31;
    int b    = row / M_;
    int base = row * K_;
    float s = __expf(logk[base + lane]      + v[b * N_ + idx[base + lane]])
            + __expf(logk[base + lane + 32] + v[b * N_ + idx[base + lane + 32]]);
#pragma unroll
    for (int off = 16; off > 0; off >>= 1) s += __shfl_down(s, off, 32);
    if (lane == 0) u[row] = (s > 0.0f) ? -__logf(s) : 0.0f;
}

// ---------------- Sinkhorn column scatter: colsum += exp(logk + u) ----------
__global__ __launch_bounds__(256) void k_scatter(
    const int* __restrict__ idx, const float* __restrict__ logk,
    const float* __restrict__ u, float* __restrict__ col)
{
    int i   = blockIdx.x * 256 + threadIdx.x;   // over B*M*64 entries
    int row = i >> 6;
    int b   = row / M_;
    float w = __expf(logk[i] + u[row]);
    if (w != 0.0f) atomicAdd(&col[b * N_ + idx[i]], w);
}

// ---------------- Sinkhorn column finish: v = -log colsum; reset colsum -----
__global__ __launch_bounds__(256) void k_finish(float* __restrict__ v,
                                                float* __restrict__ col) {
    int i = blockIdx.x * 256 + threadIdx.x;
    if (i < B_ * N_) {
        float c = col[i];
        v[i] = (c > 0.0f) ? -__logf(c) : 0.0f;
        col[i] = 0.0f;
    }
}

// ---------------- output SpMM: out[m,:] = sum_k attn_k * feats[idx_k,:] -----
__global__ __launch_bounds__(128) void k_out(
    const float* __restrict__ feats,          // [B,N,D]
    const int* __restrict__ idx, const float* __restrict__ logk,
    const float* __restrict__ u, const float* __restrict__ v,
    const int* __restrict__ cnt,
    const unsigned char* __restrict__ tmask,
    float* __restrict__ out)                  // [B,M,D]
{
    __shared__ float w[K_];
    __shared__ int   id[K_];
    int row = blockIdx.x;            // b*M + m
    int b   = row / M_;
    int tid = threadIdx.x;           // = feature dim d
    if (tid < K_) {
        int i = row * K_ + tid;
        w[tid]  = __expf(logk[i] + u[row] + v[b * N_ + idx[i]]);
        id[tid] = idx[i];
    }
    __syncthreads();
    const float* fb = feats + (size_t)b * N_ * D_;
    float acc = 0.0f;
#pragma unroll 8
    for (int k = 0; k < K_; ++k) acc += w[k] * fb[(size_t)id[k] * D_ + tid];
    float scale = (cnt[row] > 0 && tmask[row] != 0) ? 1.0f : 0.0f;
    out[(size_t)row * D_ + tid] = acc * scale;
}

// ===================================================================
extern "C" void kernel_launch(void* const* d_in, const int* in_sizes, int n_in,
                              void* d_out, int out_size, void* d_ws, size_t ws_size,
                              hipStream_t stream) {
    const float* src_feats = (const float*)d_in[0];          // [B,N,D]
    const float* src_locs  = (const float*)d_in[1];          // [B,N,2]
    const float* tgt_locs  = (const float*)d_in[2];          // [B,M,2]
    const unsigned char* smask = (const unsigned char*)d_in[3]; // [B,N]
    const unsigned char* tmask = (const unsigned char*)d_in[4]; // [B,M]
    float* out = (float*)d_out;                               // [B,M,D]

    const int BM = B_ * M_;
    const int BN = B_ * N_;

    // workspace layout (all 4-byte types, naturally aligned)
    float* ws_logk = (float*)d_ws;               // BM*64
    int*   ws_idx  = (int*)(ws_logk + BM * K_);  // BM*64
    int*   ws_cnt  = (int*)(ws_idx + BM * K_);   // BM
    float* ws_u    = (float*)(ws_cnt + BM);      // BM
    float* ws_v    = ws_u + BM;                  // BN
    float* ws_col  = ws_v + BN;                  // BN

    // 1) top-k sparse support via WMMA distance tiles
    k_topk<<<B_ * (M_ / 16), 128, 0, stream>>>(src_locs, tgt_locs, smask, tmask,
                                               ws_idx, ws_logk, ws_cnt);

    // 2) Sinkhorn: init v=0, colsum=0, then 8 x (row, scatter, finish)
    k_init<<<(BN + 255) / 256, 256, 0, stream>>>(ws_v, ws_col);
    for (int it = 0; it < IT_; ++it) {
        k_row<<<BM / 8, 256, 0, stream>>>(ws_idx, ws_logk, ws_v, ws_u);
        k_scatter<<<(BM * K_) / 256, 256, 0, stream>>>(ws_idx, ws_logk, ws_u, ws_col);
        k_finish<<<(BN + 255) / 256, 256, 0, stream>>>(ws_v, ws_col);
    }

    // 3) sparse attn x feats
    k_out<<<BM, 128, 0, stream>>>(src_feats, ws_idx, ws_logk, ws_u, ws_v,
                                  ws_cnt, tmask, out);
}